// DummyFlow_43533788512483
// MI455X (gfx1250) — compile-verified
//
#include <hip/hip_runtime.h>

// Output (128, 2, 256, 256) f32 = temporal diff of channel 0 of x(8,16,3,256,256),
// broadcast over 2 channels. Pure streaming kernel: 32 MB unique read + 64 MB write,
// HBM-bound (~4 us floor at 23.3 TB/s). float4 (B128) accesses, NT stores to keep
// the twice-read input resident in the 192 MB L2.

typedef float float4v __attribute__((ext_vector_type(4)));

__global__ void diff_broadcast_kernel(const float* __restrict__ x,
                                      float* __restrict__ out) {
    constexpr int HW   = 256 * 256;   // pixels per plane
    constexpr int HW4  = HW / 4;      // float4 per plane (16384)
    constexpr int CHW  = 3 * HW;      // x stride between (b,l) planes (channel dim = 3)
    constexpr int L    = 16;

    const unsigned idx = blockIdx.x * blockDim.x + threadIdx.x;
    const unsigned hw4 = idx & (HW4 - 1);     // float4 index within plane
    const unsigned n   = idx >> 14;           // flattened (b,l) plane, 0..127
    const unsigned l   = n & (L - 1);

    float4v d;
    if (l == 0) {
        // first frame of each sequence: diff is zero
        d = (float4v){0.0f, 0.0f, 0.0f, 0.0f};
    } else {
        // channel 0 of plane n and plane n-1 (regular temporal loads: reused, keep in L2)
        const float4v* cur  = (const float4v*)(x + (size_t)n       * CHW) + hw4;
        const float4v* prev = (const float4v*)(x + (size_t)(n - 1) * CHW) + hw4;
        d = *cur - *prev;
    }

    // broadcast to both output channels; streaming writes -> non-temporal
    float4v* o = (float4v*)out + (size_t)n * (2 * HW4) + hw4;
    __builtin_nontemporal_store(d, o);        // channel 0
    __builtin_nontemporal_store(d, o + HW4);  // channel 1
}

extern "C" void kernel_launch(void* const* d_in, const int* in_sizes, int n_in,
                              void* d_out, int out_size, void* d_ws, size_t ws_size,
                              hipStream_t stream) {
    (void)in_sizes; (void)n_in; (void)d_ws; (void)ws_size;

    const float* x = (const float*)d_in[0];
    float*     out = (float*)d_out;

    constexpr int HW4 = 256 * 256 / 4;            // 16384 float4 per plane
    const int planes  = out_size / (2 * 256 * 256); // 128
    const int threads = 256;
    const int total   = planes * HW4;             // 2,097,152 threads
    const int blocks  = total / threads;          // 8192

    diff_broadcast_kernel<<<blocks, threads, 0, stream>>>(x, out);
}